// MultiScaleProcessor_29738353557936
// MI455X (gfx1250) — compile-verified
//
#include <hip/hip_runtime.h>

typedef float v2f __attribute__((ext_vector_type(2)));
typedef float v8f __attribute__((ext_vector_type(8)));

#define NPTS 8192
#define DIM  256
#define NB   8
#define HSTR 260   // padded LDS row stride (floats) to avoid 32-way bank conflicts

// ---------------------------------------------------------------------------
// Kernel 1: furthest point sampling. grid(3 scales, 8 batches), 1024 threads.
// xyz held in LDS; per-point running min-distance in registers (8 pts/thread);
// argmax via wave32 shuffle reduction + 32-entry LDS tree. First-index tiebreak
// matches jnp.argmax.
// ---------------------------------------------------------------------------
__global__ __launch_bounds__(1024) void fps_kernel(const float* __restrict__ xyz,
                                                   int* __restrict__ idx_out)
{
  const int sc = blockIdx.x;         // 0..2  -> n_samples = 512 << sc
  const int b  = blockIdx.y;         // 0..7
  const int ns = 512 << sc;
  const int tid = threadIdx.x;

  extern __shared__ float smem[];
  float* sx = smem;
  float* sy = sx + NPTS;
  float* sz = sy + NPTS;
  float* red_d = sz + NPTS;          // 32 floats
  int*   red_i = (int*)(red_d + 32); // 32 ints

  for (int i = tid; i < NPTS; i += 1024) {
    const float* p = xyz + ((size_t)b * NPTS + i) * 3;
    sx[i] = p[0]; sy[i] = p[1]; sz[i] = p[2];
  }
  __syncthreads();

  float dist[8];
#pragma unroll
  for (int k = 0; k < 8; ++k) dist[k] = 1e10f;

  const int base = (b * 3 + sc) * 2048;
  if (tid == 0) idx_out[base] = 0;   // init_idx = 0
  int last = 0;

  for (int j = 1; j < ns; ++j) {
    const float lx = sx[last], ly = sy[last], lz = sz[last];
    float best = -1.0f; int bi = 0x7fffffff;
#pragma unroll
    for (int k = 0; k < 8; ++k) {
      const int i = tid + k * 1024;
      const float dx = sx[i] - lx, dy = sy[i] - ly, dz = sz[i] - lz;
      const float d = dx * dx + dy * dy + dz * dz;
      dist[k] = fminf(dist[k], d);
      if (dist[k] > best || (dist[k] == best && i < bi)) { best = dist[k]; bi = i; }
    }
#pragma unroll
    for (int off = 16; off >= 1; off >>= 1) {
      const float od = __shfl_xor(best, off, 32);
      const int   oi = __shfl_xor(bi,   off, 32);
      if (od > best || (od == best && oi < bi)) { best = od; bi = oi; }
    }
    const int wv = tid >> 5;
    if ((tid & 31) == 0) { red_d[wv] = best; red_i[wv] = bi; }
    __syncthreads();
    if (wv == 0) {
      best = red_d[tid]; bi = red_i[tid];     // 32 waves -> 32 entries
#pragma unroll
      for (int off = 16; off >= 1; off >>= 1) {
        const float od = __shfl_xor(best, off, 32);
        const int   oi = __shfl_xor(bi,   off, 32);
        if (od > best || (od == best && oi < bi)) { best = od; bi = oi; }
      }
      if (tid == 0) { red_i[0] = bi; idx_out[base + j] = bi; }
    }
    __syncthreads();
    last = red_i[0];
    __syncthreads();
  }
}

// ---------------------------------------------------------------------------
// Kernel 2: per-scale MLP on gathered tokens:  Linear -> LN -> ReLU -> Linear.
// One workgroup (256 thr = 8 waves) per 64-row tile. Wave w computes a 16x128
// strip as 8 accumulators of V_WMMA_F32_16X16X4_F32 (full f32 precision).
// ---------------------------------------------------------------------------
__global__ __launch_bounds__(256) void scale_mlp_kernel(
    const float* __restrict__ features, const int* __restrict__ idx_ws,
    const float* __restrict__ w1, const float* __restrict__ b1,
    const float* __restrict__ g1, const float* __restrict__ be1,
    const float* __restrict__ w2, const float* __restrict__ b2,
    float* __restrict__ proc)
{
  const int bx = blockIdx.x;         // 0..55  (8 + 16 + 32 tiles)
  const int b  = blockIdx.y;
  int sc, tile;
  if (bx < 8)       { sc = 0; tile = bx; }
  else if (bx < 24) { sc = 1; tile = bx - 8; }
  else              { sc = 2; tile = bx - 24; }
  const int s  = 512 << sc;
  const int m0 = tile * 64;

  const float* W1 = w1  + (size_t)sc * DIM * DIM;
  const float* W2 = w2  + (size_t)sc * DIM * DIM;
  const float* B1 = b1  + sc * DIM;
  const float* B2 = b2  + sc * DIM;
  const float* G  = g1  + sc * DIM;
  const float* BE = be1 + sc * DIM;
  const size_t procOff = (sc == 0) ? 0
                       : (sc == 1) ? (size_t)NB * 512 * DIM
                                   : (size_t)NB * (512 + 1024) * DIM;

  extern __shared__ float smem[];
  float* Hs   = smem;                       // 64 x HSTR
  int*   ridx = (int*)(Hs + 64 * HSTR);     // 64 gathered row indices

  const int tid = threadIdx.x;
  if (tid < 64) ridx[tid] = idx_ws[(b * 3 + sc) * 2048 + m0 + tid];
  __syncthreads();

  const int lane = tid & 31, wave = tid >> 5;
  const int mstrip = wave & 3;              // 4 strips of 16 rows
  const int nhalf  = wave >> 1 & 0 | (wave >> 2); // == wave >> 2
  const int lm = lane & 15;
  const int kh = lane >> 4;                 // A/B frag K-half select

  const v8f vzero = {};
  v8f acc[8];
#pragma unroll
  for (int nt = 0; nt < 8; ++nt) acc[nt] = vzero;

  // ---- GEMM1: gathered X @ W1  (K = 256 as 64 steps of k=4) ----
  const float* xrow = features + ((size_t)b * NPTS + ridx[mstrip * 16 + lm]) * DIM;
  for (int ks = 0; ks < 64; ++ks) {
    const int kb = ks * 4 + 2 * kh;
    v2f a; a.x = xrow[kb]; a.y = xrow[kb + 1];
    const float* wp = W1 + (size_t)kb * DIM + nhalf * 128 + lm;
#pragma unroll
    for (int nt = 0; nt < 8; ++nt) {
      v2f bf; bf.x = wp[nt * 16]; bf.y = wp[nt * 16 + DIM];
      acc[nt] = __builtin_amdgcn_wmma_f32_16x16x4_f32(false, a, false, bf,
                                                      (short)0, acc[nt], false, false);
    }
  }
#pragma unroll
  for (int nt = 0; nt < 8; ++nt) {
    const int n = nhalf * 128 + nt * 16 + lm;
    const float bias = B1[n];
#pragma unroll
    for (int r = 0; r < 8; ++r)
      Hs[(mstrip * 16 + r + 8 * kh) * HSTR + n] = acc[nt][r] + bias;
  }
  __syncthreads();

  // ---- LayerNorm + ReLU over LDS rows ----
  if (tid < 64) {
    float* row = Hs + tid * HSTR;
    float mu = 0.f;
    for (int i = 0; i < DIM; ++i) mu += row[i];
    mu *= (1.0f / DIM);
    float var = 0.f;
    for (int i = 0; i < DIM; ++i) { const float d = row[i] - mu; var += d * d; }
    var *= (1.0f / DIM);
    const float rstd = rsqrtf(var + 1e-5f);
    for (int i = 0; i < DIM; ++i) {
      const float v = (row[i] - mu) * rstd * G[i] + BE[i];
      row[i] = fmaxf(v, 0.0f);
    }
  }
  __syncthreads();

  // ---- GEMM2: H @ W2 + b2 -> proc ----
#pragma unroll
  for (int nt = 0; nt < 8; ++nt) acc[nt] = vzero;
  const float* hrow = Hs + (mstrip * 16 + lm) * HSTR;
  for (int ks = 0; ks < 64; ++ks) {
    const int kb = ks * 4 + 2 * kh;
    v2f a; a.x = hrow[kb]; a.y = hrow[kb + 1];
    const float* wp = W2 + (size_t)kb * DIM + nhalf * 128 + lm;
#pragma unroll
    for (int nt = 0; nt < 8; ++nt) {
      v2f bf; bf.x = wp[nt * 16]; bf.y = wp[nt * 16 + DIM];
      acc[nt] = __builtin_amdgcn_wmma_f32_16x16x4_f32(false, a, false, bf,
                                                      (short)0, acc[nt], false, false);
    }
  }
  float* prow = proc + procOff + ((size_t)b * s + m0) * DIM;
#pragma unroll
  for (int nt = 0; nt < 8; ++nt) {
    const int n = nhalf * 128 + nt * 16 + lm;
    const float bias = B2[n];
#pragma unroll
    for (int r = 0; r < 8; ++r)
      prow[(size_t)(mstrip * 16 + r + 8 * kh) * DIM + n] = acc[nt][r] + bias;
  }
}

// ---------------------------------------------------------------------------
// Kernel 3: fusion MLP. The reference tiles proc by n % s, so the 768-wide cat
// rows repeat with period lcm(512,1024,2048)=2048 -> compute only 2048 unique
// rows per batch (4x less work), then broadcast to 4 output rows each.
// ---------------------------------------------------------------------------
__global__ __launch_bounds__(256) void fusion_kernel(
    const float* __restrict__ proc,
    const float* __restrict__ fw1, const float* __restrict__ fb1,
    const float* __restrict__ fg,  const float* __restrict__ fbe,
    const float* __restrict__ fw2, const float* __restrict__ fb2,
    float* __restrict__ out)
{
  const int tile = blockIdx.x;       // 0..31 (2048 unique rows / 64)
  const int b    = blockIdx.y;
  const int u0   = tile * 64;

  extern __shared__ float smem[];
  float* Hs = smem;                  // 64 x HSTR

  const int tid = threadIdx.x;
  const int lane = tid & 31, wave = tid >> 5;
  const int mstrip = wave & 3, nhalf = wave >> 2;
  const int lm = lane & 15, kh = lane >> 4;

  const float* proc0 = proc;
  const float* proc1 = proc + (size_t)NB * 512 * DIM;
  const float* proc2 = proc + (size_t)NB * (512 + 1024) * DIM;

  const int u = u0 + mstrip * 16 + lm;
  const float* rowp[3] = {
    proc0 + ((size_t)b * 512  + (u & 511))  * DIM,
    proc1 + ((size_t)b * 1024 + (u & 1023)) * DIM,
    proc2 + ((size_t)b * 2048 + u)          * DIM };

  const v8f vzero = {};
  v8f acc[8];
#pragma unroll
  for (int nt = 0; nt < 8; ++nt) acc[nt] = vzero;

  // ---- GEMM1: cat(768) @ fw1 ; K split over the 3 proc segments ----
#pragma unroll
  for (int seg = 0; seg < 3; ++seg) {
    const float* rp   = rowp[seg];
    const float* wseg = fw1 + (size_t)seg * 256 * DIM;
    for (int ks = 0; ks < 64; ++ks) {
      const int kb = ks * 4 + 2 * kh;
      v2f a; a.x = rp[kb]; a.y = rp[kb + 1];
      const float* wp = wseg + (size_t)kb * DIM + nhalf * 128 + lm;
#pragma unroll
      for (int nt = 0; nt < 8; ++nt) {
        v2f bf; bf.x = wp[nt * 16]; bf.y = wp[nt * 16 + DIM];
        acc[nt] = __builtin_amdgcn_wmma_f32_16x16x4_f32(false, a, false, bf,
                                                        (short)0, acc[nt], false, false);
      }
    }
  }
#pragma unroll
  for (int nt = 0; nt < 8; ++nt) {
    const int n = nhalf * 128 + nt * 16 + lm;
    const float bias = fb1[n];
#pragma unroll
    for (int r = 0; r < 8; ++r)
      Hs[(mstrip * 16 + r + 8 * kh) * HSTR + n] = acc[nt][r] + bias;
  }
  __syncthreads();

  if (tid < 64) {
    float* row = Hs + tid * HSTR;
    float mu = 0.f;
    for (int i = 0; i < DIM; ++i) mu += row[i];
    mu *= (1.0f / DIM);
    float var = 0.f;
    for (int i = 0; i < DIM; ++i) { const float d = row[i] - mu; var += d * d; }
    var *= (1.0f / DIM);
    const float rstd = rsqrtf(var + 1e-5f);
    for (int i = 0; i < DIM; ++i) {
      const float v = (row[i] - mu) * rstd * fg[i] + fbe[i];
      row[i] = fmaxf(v, 0.0f);
    }
  }
  __syncthreads();

  // ---- GEMM2 + 4-way replicated store ----
#pragma unroll
  for (int nt = 0; nt < 8; ++nt) acc[nt] = vzero;
  const float* hrow = Hs + (mstrip * 16 + lm) * HSTR;
  for (int ks = 0; ks < 64; ++ks) {
    const int kb = ks * 4 + 2 * kh;
    v2f a; a.x = hrow[kb]; a.y = hrow[kb + 1];
    const float* wp = fw2 + (size_t)kb * DIM + nhalf * 128 + lm;
#pragma unroll
    for (int nt = 0; nt < 8; ++nt) {
      v2f bf; bf.x = wp[nt * 16]; bf.y = wp[nt * 16 + DIM];
      acc[nt] = __builtin_amdgcn_wmma_f32_16x16x4_f32(false, a, false, bf,
                                                      (short)0, acc[nt], false, false);
    }
  }
  float* ob = out + (size_t)b * NPTS * DIM;
#pragma unroll
  for (int nt = 0; nt < 8; ++nt) {
    const int n = nhalf * 128 + nt * 16 + lm;
    const float bias = fb2[n];
#pragma unroll
    for (int r = 0; r < 8; ++r) {
      const int uu = u0 + mstrip * 16 + r + 8 * kh;
      const float v = acc[nt][r] + bias;
      ob[(size_t)(uu       ) * DIM + n] = v;
      ob[(size_t)(uu + 2048) * DIM + n] = v;
      ob[(size_t)(uu + 4096) * DIM + n] = v;
      ob[(size_t)(uu + 6144) * DIM + n] = v;
    }
  }
}

// ---------------------------------------------------------------------------
extern "C" void kernel_launch(void* const* d_in, const int* in_sizes, int n_in,
                              void* d_out, int out_size, void* d_ws, size_t ws_size,
                              hipStream_t stream)
{
  (void)in_sizes; (void)n_in; (void)out_size; (void)ws_size;

  const float* features = (const float*)d_in[0];   // [8,8192,256]
  const float* xyz      = (const float*)d_in[1];   // [8,8192,3]
  const float* w1  = (const float*)d_in[2];
  const float* b1  = (const float*)d_in[3];
  const float* g1  = (const float*)d_in[4];
  const float* be1 = (const float*)d_in[5];
  const float* w2  = (const float*)d_in[6];
  const float* b2  = (const float*)d_in[7];
  const float* fw1 = (const float*)d_in[8];
  const float* fb1 = (const float*)d_in[9];
  const float* fg  = (const float*)d_in[10];
  const float* fbe = (const float*)d_in[11];
  const float* fw2 = (const float*)d_in[12];
  const float* fb2 = (const float*)d_in[13];
  float* out = (float*)d_out;

  // workspace: [8][3][2048] int indices, then f32 proc buffers
  //   proc0 [8][512][256] ++ proc1 [8][1024][256] ++ proc2 [8][2048][256]
  int*   idx_ws = (int*)d_ws;
  float* proc   = (float*)((char*)d_ws + (size_t)NB * 3 * 2048 * sizeof(int));

  const size_t fps_lds = (size_t)(3 * NPTS + 32) * sizeof(float) + 32 * sizeof(int);
  fps_kernel<<<dim3(3, NB), 1024, fps_lds, stream>>>(xyz, idx_ws);

  const size_t mlp_lds = (size_t)64 * HSTR * sizeof(float) + 64 * sizeof(int);
  scale_mlp_kernel<<<dim3(56, NB), 256, mlp_lds, stream>>>(
      features, idx_ws, w1, b1, g1, be1, w2, b2, proc);

  const size_t fus_lds = (size_t)64 * HSTR * sizeof(float);
  fusion_kernel<<<dim3(32, NB), 256, fus_lds, stream>>>(
      proc, fw1, fb1, fg, fbe, fw2, fb2, out);
}